// ThinVesselLoss_52278341927515
// MI455X (gfx1250) — compile-verified
//
#include <hip/hip_runtime.h>
#include <hip/hip_bf16.h>
#include <math.h>

// Problem constants (B,C,H,W) = (4,1,512,512) from the reference setup.
#define IMGS   4
#define HH     512
#define WW     512
#define NTOT   (IMGS * HH * WW)          // 2^20 elements
#define TILE   16
#define HALO   (TILE + 2)                // 18x18 LDS tile with 1-px halo
#define NBLOCKS ((WW / TILE) * (HH / TILE) * IMGS)  // 32*32*4 = 4096

typedef __attribute__((ext_vector_type(2))) float v2f;
typedef __attribute__((ext_vector_type(8))) float v8f;

// Wave32 sum reduction routed through the matrix pipe.
// D = ones(16x4) x B(4x16): every row of D equals the column sums of B.
// With every B slot = this lane's scalar s, each lane's value is counted
// twice in the column sums, and D's first VGPR covers each of the 16
// columns exactly twice across the 32 lanes. Hence
//   sum_over_lanes(D.vgpr0) = 4 * sum_over_lanes(s),
// independent of the precise B lane<->element mapping.
__device__ __forceinline__ float wave_sum_wmma(float s) {
    v2f a; a.x = 1.0f; a.y = 1.0f;   // A = ones (16x4 f32, 2 VGPRs/lane)
    v2f b; b.x = s;    b.y = s;      // every B slot = lane scalar
    v8f c = {};
    c = __builtin_amdgcn_wmma_f32_16x16x4_f32(
        /*neg_a=*/false, a, /*neg_b=*/false, b,
        /*c_mod=*/(short)0, c, /*reuse_a=*/false, /*reuse_b=*/false);
    float d = c[0];
    d += __shfl_xor(d, 16, 32);
    d += __shfl_xor(d,  8, 32);
    d += __shfl_xor(d,  4, 32);
    d += __shfl_xor(d,  2, 32);
    d += __shfl_xor(d,  1, 32);
    return d * 0.25f;                // exact scaling (power of two)
}

__global__ __launch_bounds__(256)
void ThinVesselLoss_main(const float* __restrict__ pred,
                         const float* __restrict__ target,
                         float* __restrict__ partials) {
    __shared__ float tile[HALO * HALO];
    __shared__ float wsum[8];

    const int bx = blockIdx.x, by = blockIdx.y, img = blockIdx.z;
    const int tx = threadIdx.x, ty = threadIdx.y;
    const int tid = ty * TILE + tx;

    const float* timg = target + (size_t)img * HH * WW;

    // Cooperative halo load: 324 elements, <=2 per thread.
    // Out-of-bounds filled with 1.0 (> 0.5): counts as foreground, i.e. the
    // image border is NOT background (matches the EDT's big-sentinel seed).
    for (int idx = tid; idx < HALO * HALO; idx += 256) {
        const int ly = idx / HALO, lx = idx % HALO;
        const int gy = by * TILE + ly - 1;
        const int gx = bx * TILE + lx - 1;
        float v = 1.0f;
        if (gy >= 0 && gy < HH && gx >= 0 && gx < WW) v = timg[gy * WW + gx];
        tile[idx] = v;
    }
    __syncthreads();

    const int gy = by * TILE + ty;
    const int gx = bx * TILE + tx;
    const size_t gidx = (size_t)img * HH * WW + (size_t)gy * WW + gx;

    const int cy = ty + 1, cx = tx + 1;
    const float t = tile[cy * HALO + cx];
    const float p = pred[gidx];

    // thin <=> foreground with a background pixel among the 8 in-bounds
    // neighbors  (<=> 0 < EDT dist < 1.5  <=> 0 < width < 3.0)
    const bool fg = t > 0.5f;
    const bool bgnb =
        (tile[(cy - 1) * HALO + (cx - 1)] <= 0.5f) |
        (tile[(cy - 1) * HALO + (cx    )] <= 0.5f) |
        (tile[(cy - 1) * HALO + (cx + 1)] <= 0.5f) |
        (tile[(cy    ) * HALO + (cx - 1)] <= 0.5f) |
        (tile[(cy    ) * HALO + (cx + 1)] <= 0.5f) |
        (tile[(cy + 1) * HALO + (cx - 1)] <= 0.5f) |
        (tile[(cy + 1) * HALO + (cx    )] <= 0.5f) |
        (tile[(cy + 1) * HALO + (cx + 1)] <= 0.5f);
    const float w = (fg && bgnb) ? 3.0f : 1.0f;

    // BCE with torch-style log clamp at -100 (clamp inactive for p in
    // (0.001, 0.999) but kept for fidelity).
    const float lp  = fmaxf(logf(p),     -100.0f);
    const float l1p = fmaxf(log1pf(-p),  -100.0f);
    const float bce = -(t * lp + (1.0f - t) * l1p);
    const float s = bce * w;

    // Deterministic block reduction: WMMA wave sum -> LDS -> thread 0.
    const float wred = wave_sum_wmma(s);
    const int wave = tid >> 5, lane = tid & 31;
    if (lane == 0) wsum[wave] = wred;
    __syncthreads();
    if (tid == 0) {
        float tot = 0.0f;
        for (int i = 0; i < 8; ++i) tot += wsum[i];
        partials[((size_t)img * gridDim.y + by) * gridDim.x + bx] = tot;
    }
}

__global__ __launch_bounds__(256)
void ThinVesselLoss_reduce(const float* __restrict__ partials,
                           float* __restrict__ out) {
    __shared__ float wsum[8];
    const int tid = threadIdx.x;

    // Fixed-order strided accumulation: deterministic across replays.
    float s = 0.0f;
    for (int i = tid; i < NBLOCKS; i += 256) s += partials[i];

    const float wred = wave_sum_wmma(s);
    const int wave = tid >> 5, lane = tid & 31;
    if (lane == 0) wsum[wave] = wred;
    __syncthreads();
    if (tid == 0) {
        float tot = 0.0f;
        for (int i = 0; i < 8; ++i) tot += wsum[i];
        out[0] = tot * (1.0f / (float)NTOT);  // exact: NTOT = 2^20
    }
}

extern "C" void kernel_launch(void* const* d_in, const int* in_sizes, int n_in,
                              void* d_out, int out_size, void* d_ws, size_t ws_size,
                              hipStream_t stream) {
    const float* pred   = (const float*)d_in[0];   // [4,1,512,512] f32
    const float* target = (const float*)d_in[1];   // [4,1,512,512] f32
    float* out      = (float*)d_out;               // scalar f32
    float* partials = (float*)d_ws;                // 4096 f32 = 16 KB scratch

    dim3 grid(WW / TILE, HH / TILE, IMGS);         // (32, 32, 4)
    dim3 block(TILE, TILE);                        // 256 threads = 8 waves
    ThinVesselLoss_main<<<grid, block, 0, stream>>>(pred, target, partials);
    ThinVesselLoss_reduce<<<1, 256, 0, stream>>>(partials, out);
}